// WaveletEmbeding_9672266350697
// MI455X (gfx1250) — compile-verified
//
#include <hip/hip_runtime.h>
#include <hip/hip_bf16.h>
#include <stdint.h>

typedef __attribute__((ext_vector_type(2))) float v2f;
typedef __attribute__((ext_vector_type(8))) float v8f;
typedef __attribute__((ext_vector_type(4))) int   v4i;

#define NB   32      // batch
#define NC   8       // channels
#define NL   512     // sequence length
#define NS   63      // wavelet scales
#define CTOT 512     // C*(1+S) = 8*64
#define DMOD 512     // d_model

// ---- CDNA5 async global->LDS path (ASYNCcnt), guarded so compile never regresses
#if defined(__has_builtin)
# if __has_builtin(__builtin_amdgcn_global_load_async_to_lds_b128)
#  define USE_ASYNC_LDS 1
# endif
#endif
#ifndef USE_ASYNC_LDS
# define USE_ASYNC_LDS 0
#endif

// Builtin expects pointers to 16-byte int vectors (per clang diagnostic):
typedef __attribute__((address_space(1))) v4i* gv4p_t;   // global src
typedef __attribute__((address_space(3))) v4i* lv4p_t;   // LDS dst

static __device__ __forceinline__ void wait_asynccnt0() {
#if defined(__has_builtin) && __has_builtin(__builtin_amdgcn_s_wait_asynccnt)
  __builtin_amdgcn_s_wait_asynccnt(0);
#else
  asm volatile("s_wait_asynccnt 0x0" ::: "memory");
#endif
}

// ---------------------------------------------------------------------------
// Kernel 1: fused CWT (per-scale correlation) + |diff| + concat.
// Produces ws[b][ct][l] (row-major f32); ct 0..7 = x rows, ct = 8+c*63+s =
// wavelet features.  One block per (bc, j); j=0 copies the x row, j=1..63 is
// scale s=j-1.  Uses the exact 16*(s+1)+1 tap support of each zero-padded kernel
// row (halves the FMA count vs dense K=1009).  VALU is the right tool here
// (irregular taps + boundary clipping), WMMA handles the tokenConv GEMM.
// ---------------------------------------------------------------------------
__global__ __launch_bounds__(256) void cwt_diff_kernel(
    const float* __restrict__ x,      // [256][512]
    const float* __restrict__ bank,   // [63][K]
    float* __restrict__ outw,         // [32][512][512]
    int K) {
  __shared__ float xs[NL];
  __shared__ float cv[NL];
  __shared__ float bk[1024];          // K = 1009 <= 1024

  const int blk = blockIdx.x;
  const int bc  = blk >> 6;           // 0..255
  const int j   = blk & 63;           // 0 = copy, 1..63 = scale j-1
  const int b   = bc >> 3;
  const int c   = bc & 7;
  const int tid = threadIdx.x;

  const float* xrow = x + (size_t)bc * NL;

  if (j == 0) {                       // pass-through rows (concat head)
    float* dst = outw + ((size_t)b * CTOT + c) * NL;
    dst[tid]       = xrow[tid];
    dst[tid + 256] = xrow[tid + 256];
    return;
  }

  xs[tid]       = xrow[tid];
  xs[tid + 256] = xrow[tid + 256];

  const int s   = j - 1;              // scale-1
  const int len = 16 * (s + 1) + 1;   // non-zero taps of this row
  const int off = (K - len) >> 1;     // centered zero padding
  const int pad = (K - 1) >> 1;

  const float* brow = bank + (size_t)s * K;
  for (int k = tid; k < K && k < 1024; k += 256) bk[k] = brow[k];
  __syncthreads();

  // conv[p] = sum_k bk[k] * x[p + k - pad], x zero outside [0,512)
#pragma unroll
  for (int rep = 0; rep < 2; ++rep) {
    const int p  = tid + rep * 256;
    int kb = off;           if (pad - p > kb)        kb = pad - p;
    int ke = off + len;     if (512 + pad - p < ke)  ke = 512 + pad - p;
    float acc = 0.0f;
    int xi = p + kb - pad;
    for (int k = kb; k < ke; ++k, ++xi) acc = fmaf(bk[k], xs[xi], acc);
    cv[p] = acc;
  }
  __syncthreads();

  float* drow = outw + ((size_t)b * CTOT + (8 + c * NS + s)) * NL;
#pragma unroll
  for (int rep = 0; rep < 2; ++rep) {
    const int p  = tid + rep * 256;
    const int lp = (p > 510) ? 510 : p;           // replicate last diff
    drow[p] = fabsf(cv[lp + 1] - cv[lp]);
  }
}

// ---------------------------------------------------------------------------
// Kernel 2: tokenConv as batched WMMA GEMM with 3-tap circular row shift:
//   ret[b, ct, m] = sum_{k=0..2} sum_l ws[b, (ct+k-1)&511, l] * W[m, l, k]
// 64x64 output tile / 256-thread block; grid (8 Mtiles, 8 Ntiles, 32 batch).
// Double-buffered LDS staging via GLOBAL_LOAD_ASYNC_TO_LDS_B128 (ASYNCcnt):
// chunk i+1 streams into LDS while WMMAs consume chunk i.
//   As rows padded to 20 floats (80B: b128-aligned, 20*k mod 64 distinct)
//   Ws rows padded to 52 floats (208B: b128-aligned, 52*k mod 64 distinct)
// f32 WMMA 16x16x4: compute is tiny vs the HBM roofline -> keep full f32.
// ---------------------------------------------------------------------------
__global__ __launch_bounds__(256) void tokenconv_wmma_kernel(
    const float* __restrict__ outw,   // [32][512][512]
    const float* __restrict__ W,      // [512][512][3]
    float* __restrict__ y) {          // [32][512][512]
  __shared__ float As[2][66][20];     // A window: 66 shifted rows x 16 K-cols
  __shared__ float Ws[2][64][52];     // raw W tile: 64 cols x 48 floats (16 l x 3 taps)

  const int mt = blockIdx.x, nt = blockIdx.y, b = blockIdx.z;
  const int Mbase = mt * 64, Nbase = nt * 64;
  const int tid  = threadIdx.x;
  const int wave = tid >> 5;
  const int lane = tid & 31;
  const int half = lane >> 4;         // 0: lanes 0-15, 1: lanes 16-31
  const int lid  = lane & 15;
  const int tm   = wave & 3;          // M sub-tile (4)
  const int tn0  = (wave >> 2) * 2;   // first of two N sub-tiles

  v8f acc0 = {};
  v8f acc1 = {};

  const float* Ab = outw + (size_t)b * CTOT * NL;

  // Stage one 16-wide K-chunk into LDS buffer `buf`.
  auto stage = [&](int buf, int Lb) {
#if USE_ASYNC_LDS
    // A window: 66 rows x 16 floats = 264 b128 transfers
    for (int i = tid; i < 264; i += 256) {
      const int r  = i >> 2;
      const int cc = (i & 3) << 2;
      const int ct = (Mbase - 1 + r) & 511;
      const float* g = Ab + (size_t)ct * NL + Lb + cc;
      __builtin_amdgcn_global_load_async_to_lds_b128(
          (gv4p_t)(uintptr_t)g, (lv4p_t)(uintptr_t)&As[buf][r][cc], 0, 0);
    }
    // W tile: 64 cols x 48 floats = 768 b128 transfers (rows 16B-aligned)
    for (int i = tid; i < 768; i += 256) {
      const int n  = i / 12;
      const int cc = (i - n * 12) << 2;
      const float* g = W + (size_t)(Nbase + n) * 1536 + Lb * 3 + cc;
      __builtin_amdgcn_global_load_async_to_lds_b128(
          (gv4p_t)(uintptr_t)g, (lv4p_t)(uintptr_t)&Ws[buf][n][cc], 0, 0);
    }
#else
    for (int i = tid; i < 66 * 16; i += 256) {
      const int r  = i >> 4;
      const int cc = i & 15;
      const int ct = (Mbase - 1 + r) & 511;
      As[buf][r][cc] = Ab[(size_t)ct * NL + Lb + cc];
    }
    for (int i = tid; i < 3072; i += 256) {
      const int n   = i / 48;
      const int rem = i - n * 48;
      Ws[buf][n][rem] = W[(size_t)(Nbase + n) * 1536 + Lb * 3 + rem];
    }
#endif
  };

  stage(0, 0);
  wait_asynccnt0();
  __syncthreads();

  for (int ch = 0; ch < 32; ++ch) {
    const int buf = ch & 1;
    if (ch + 1 < 32) stage(buf ^ 1, (ch + 1) * 16);   // async prefetch next chunk

    // ---- 3 shifts x 4 K-steps of f32 16x16x4 WMMA per 16x16 sub-tile
    const int nb0 = tn0 * 16 + lid;
    const int nb1 = nb0 + 16;
    for (int sh = 0; sh < 3; ++sh) {
#pragma unroll
      for (int kk = 0; kk < 4; ++kk) {
        const int kcol = kk * 4 + half * 2;      // A/B K layout (ISA 7.12.2)
        const int ar   = tm * 16 + lid + sh;     // row ct+sh-1 of this tile row
        v2f a;
        a.x = As[buf][ar][kcol];
        a.y = As[buf][ar][kcol + 1];
        v2f b0, b1;
        b0.x = Ws[buf][nb0][kcol * 3 + sh];      // B[l, n] = W[n, l, sh]
        b0.y = Ws[buf][nb0][kcol * 3 + 3 + sh];
        b1.x = Ws[buf][nb1][kcol * 3 + sh];
        b1.y = Ws[buf][nb1][kcol * 3 + 3 + sh];
        acc0 = __builtin_amdgcn_wmma_f32_16x16x4_f32(
            false, a, false, b0, (short)0, acc0, false, false);
        acc1 = __builtin_amdgcn_wmma_f32_16x16x4_f32(
            false, a, false, b1, (short)0, acc1, false, false);
      }
    }

    wait_asynccnt0();     // own async loads for chunk ch+1 have landed
    __syncthreads();      // all waves done computing on `buf` + all loads visible
  }

  // ---- write back: C/D layout = VGPR v -> row v + 8*half, lane -> col
  float* yb = y + (size_t)b * CTOT * DMOD;
#pragma unroll
  for (int v = 0; v < 8; ++v) {
    const int row = Mbase + tm * 16 + v + half * 8;
    yb[(size_t)row * DMOD + Nbase + tn0 * 16 + lid]      = acc0[v];
    yb[(size_t)row * DMOD + Nbase + tn0 * 16 + 16 + lid] = acc1[v];
  }
}

// ---------------------------------------------------------------------------
extern "C" void kernel_launch(void* const* d_in, const int* in_sizes, int n_in,
                              void* d_out, int out_size, void* d_ws, size_t ws_size,
                              hipStream_t stream) {
  const float* x    = (const float*)d_in[0];   // [32,8,512]
  const float* bank = (const float*)d_in[1];   // [63,K]
  const float* W    = (const float*)d_in[2];   // [512,512,3]
  float* out = (float*)d_out;                  // [32,512,512]
  float* ws  = (float*)d_ws;                   // 32 MiB intermediate

  const int K = in_sizes[1] / NS;              // 1009

  // Phase 1: CWT + |diff| + concat  -> ws[32][512][512]
  cwt_diff_kernel<<<dim3(256 * 64), dim3(256), 0, stream>>>(x, bank, ws, K);

  // Phase 2: batched shifted GEMM (WMMA f32, async double-buffered LDS)
  tokenconv_wmma_kernel<<<dim3(8, 8, 32), dim3(256), 0, stream>>>(ws, W, out);
}